// PositionalScore_13005160972517
// MI455X (gfx1250) — compile-verified
//
#include <hip/hip_runtime.h>

// Problem dimensions (from reference)
constexpr int Lc = 8192;    // embedding rows
constexpr int Dc = 64;      // embedding dim
constexpr int Tc = 9;       // number of bilinear terms
constexpr int Pc = 131072;  // pairs per term
constexpr int Qc = 65536;   // loop-table lookups

typedef __attribute__((ext_vector_type(2))) float v2f;
typedef __attribute__((ext_vector_type(8))) float v8f;

// ---------------------------------------------------------------------------
// Kernel 1: Z[t] = embedding @ W[t]^T via V_WMMA_F32_16X16X4_F32.
// Z[t][j][d] = sum_e emb[j][e] * W[t][d][e]
// One wave per 16x16 output tile, K-loop of 16 WMMA ops (K=4 each).
// A (16x4 f32): lane m=lane&15 holds row m; v0/v1 hold K = kh*2, kh*2+1
//   (kh = lane>>4). B (4x16) mirrors with lane = column n.
// C/D (16x16 f32, 8 VGPRs): vr -> (M=r, N=lane) lanes 0-15, (M=r+8) lanes 16-31.
// ---------------------------------------------------------------------------
__global__ __launch_bounds__(256) void zgemm_wmma(const float* __restrict__ emb,
                                                  const float* __restrict__ W,
                                                  float* __restrict__ Z)
{
    const int wave = (blockIdx.x * blockDim.x + threadIdx.x) >> 5;
    const int lane = threadIdx.x & 31;

    // tiles: t in [0,9) x jt in [0,512) x dt in [0,4)
    const int t  = wave / (512 * 4);
    const int r  = wave - t * (512 * 4);
    const int jt = r >> 2;
    const int dt = r & 3;
    const int j0 = jt * 16;
    const int d0 = dt * 16;

    const int m  = lane & 15;   // row (A) / col (B)
    const int kh = lane >> 4;   // K half: 0 -> K{0,1}, 1 -> K{2,3}

    const float* ap = emb + (j0 + m) * Dc + kh * 2;             // A: emb rows
    const float* bp = W + t * Dc * Dc + (d0 + m) * Dc + kh * 2; // B: W^T cols

    v8f acc = {};
#pragma unroll
    for (int e = 0; e < Dc; e += 4) {
        v2f a, b;
        a.x = ap[e]; a.y = ap[e + 1];
        b.x = bp[e]; b.y = bp[e + 1];
        acc = __builtin_amdgcn_wmma_f32_16x16x4_f32(
            /*neg_a=*/false, a, /*neg_b=*/false, b,
            /*c_mod=*/(short)0, acc, /*reuse_a=*/false, /*reuse_b=*/false);
    }

    // Store D: VGPR rr holds rows j0 + rr + kh*8, column d0 + m.
    float* zp = Z + (t * Lc + j0 + kh * 8) * Dc + d0 + m;
#pragma unroll
    for (int rr = 0; rr < 8; ++rr)
        zp[rr * Dc] = acc[rr];
}

// ---------------------------------------------------------------------------
// Kernel 2: per-pair dot products. 8 lanes cooperate per pair; each lane
// loads 2x float4 (32B contiguous) from emb[i] and Z[t][j] -> coalesced
// 128-bit loads per pair row. Full-wave reduction (everything sums into the
// global total anyway), then block partial written to workspace (no atomics).
// ---------------------------------------------------------------------------
__global__ __launch_bounds__(256) void pair_dot(const float* __restrict__ emb,
                                                const float* __restrict__ Z,
                                                const int* __restrict__ pidx,
                                                float* __restrict__ partials)
{
    const int tid = blockIdx.x * 256 + threadIdx.x;
    const int g   = tid >> 3;       // global pair slot in [0, T*P)
    const int sub = tid & 7;        // 8 floats of D=64 per lane

    const int t = g / Pc;
    const int i = pidx[2 * g];
    const int j = pidx[2 * g + 1];

    const float4* a = (const float4*)(emb + i * Dc + sub * 8);
    const float4* c = (const float4*)(Z + (t * Lc + j) * Dc + sub * 8);
    float4 a0 = a[0], a1 = a[1];
    float4 c0 = c[0], c1 = c[1];

    float s = a0.x * c0.x + a0.y * c0.y + a0.z * c0.z + a0.w * c0.w
            + a1.x * c1.x + a1.y * c1.y + a1.z * c1.z + a1.w * c1.w;

    // wave32 butterfly reduction
#pragma unroll
    for (int off = 16; off > 0; off >>= 1)
        s += __shfl_xor(s, off, 32);

    __shared__ float ls[8];
    if ((threadIdx.x & 31) == 0) ls[threadIdx.x >> 5] = s;
    __syncthreads();
    if (threadIdx.x == 0) {
        float bsum = 0.f;
#pragma unroll
        for (int w = 0; w < 8; ++w) bsum += ls[w];
        partials[blockIdx.x] = bsum;
    }
}

// ---------------------------------------------------------------------------
// Kernel 3: clamped length-table lookups (tables are tiny -> cache resident).
// ---------------------------------------------------------------------------
__global__ __launch_bounds__(256) void table_sum(
    const int* __restrict__ hp, const int* __restrict__ bg,
    const int* __restrict__ il, const int* __restrict__ ex,
    const int* __restrict__ sy, const int* __restrict__ as_,
    const int* __restrict__ hx,
    const float* __restrict__ hairpin, const float* __restrict__ bulge,
    const float* __restrict__ internal_len, const float* __restrict__ expl,
    const float* __restrict__ symm, const float* __restrict__ asym,
    const float* __restrict__ helix,
    float* __restrict__ partials)
{
    const int q = blockIdx.x * 256 + threadIdx.x;

    float s = hairpin[min(hp[q], 30)]
            + bulge[min(bg[q], 30)]
            + internal_len[min(il[q], 30)]
            + expl[min(ex[2 * q], 4) * 5 + min(ex[2 * q + 1], 4)]
            + symm[min(sy[q], 15)]
            + asym[min(as_[q], 28)]
            + helix[min(hx[q], 30)];

    __shared__ float ls[256];
    ls[threadIdx.x] = s;
    __syncthreads();
#pragma unroll
    for (int off = 128; off > 0; off >>= 1) {
        if (threadIdx.x < off) ls[threadIdx.x] += ls[threadIdx.x + off];
        __syncthreads();
    }
    if (threadIdx.x == 0) partials[blockIdx.x] = ls[0];
}

// ---------------------------------------------------------------------------
// Kernel 4: deterministic final reduction (fixed summation order every call)
// plus b.sum() * P bias term.
// ---------------------------------------------------------------------------
__global__ __launch_bounds__(256) void finalize(const float* __restrict__ partA, int nA,
                                                const float* __restrict__ partB, int nB,
                                                const float* __restrict__ b,
                                                float* __restrict__ out)
{
    float s = 0.f;
    for (int i = threadIdx.x; i < nA; i += 256) s += partA[i];
    for (int i = threadIdx.x; i < nB; i += 256) s += partB[i];

    __shared__ float ls[256];
    ls[threadIdx.x] = s;
    __syncthreads();
#pragma unroll
    for (int off = 128; off > 0; off >>= 1) {
        if (threadIdx.x < off) ls[threadIdx.x] += ls[threadIdx.x + off];
        __syncthreads();
    }
    if (threadIdx.x == 0) {
        float bs = 0.f;
#pragma unroll
        for (int t = 0; t < Tc; ++t) bs += b[t];
        out[0] = ls[0] + bs * (float)Pc;
    }
}

extern "C" void kernel_launch(void* const* d_in, const int* in_sizes, int n_in,
                              void* d_out, int out_size, void* d_ws, size_t ws_size,
                              hipStream_t stream)
{
    const float* emb          = (const float*)d_in[0];
    const float* W            = (const float*)d_in[1];
    const float* b            = (const float*)d_in[2];
    const float* hairpin      = (const float*)d_in[3];
    const float* bulge        = (const float*)d_in[4];
    const float* internal_len = (const float*)d_in[5];
    const float* expl         = (const float*)d_in[6];
    const float* symm         = (const float*)d_in[7];
    const float* asym         = (const float*)d_in[8];
    const float* helix        = (const float*)d_in[9];
    const int*   pair         = (const int*)d_in[10];
    const int*   hp           = (const int*)d_in[11];
    const int*   bg           = (const int*)d_in[12];
    const int*   il           = (const int*)d_in[13];
    const int*   ex           = (const int*)d_in[14];
    const int*   sy           = (const int*)d_in[15];
    const int*   as_          = (const int*)d_in[16];
    const int*   hx           = (const int*)d_in[17];

    // Workspace layout: Z [T*L*D floats] | partA [36864] | partB [256]
    float*  Z      = (float*)d_ws;
    size_t  zElems = (size_t)Tc * Lc * Dc;          // 4,718,592 floats (18.9 MB)
    float*  partA  = Z + zElems;
    const int nA   = (Tc * Pc * 8) / 256;            // 36864 blocks
    float*  partB  = partA + nA;
    const int nB   = Qc / 256;                       // 256 blocks

    // 1) Z[t] = emb @ W[t]^T  (WMMA f32 path); 18432 waves, 8 waves/block
    zgemm_wmma<<<dim3((Tc * 512 * 4) / 8), dim3(256), 0, stream>>>(emb, W, Z);

    // 2) per-pair dots -> block partials
    pair_dot<<<dim3(nA), dim3(256), 0, stream>>>(emb, Z, pair, partA);

    // 3) table lookups -> block partials
    table_sum<<<dim3(nB), dim3(256), 0, stream>>>(hp, bg, il, ex, sy, as_, hx,
                                                  hairpin, bulge, internal_len,
                                                  expl, symm, asym, helix, partB);

    // 4) deterministic final reduction + bias
    finalize<<<dim3(1), dim3(256), 0, stream>>>(partA, nA, partB, nB, b,
                                                (float*)d_out);
}